// ATSSAssigner_65919158059560
// MI455X (gfx1250) — compile-verified
//
#include <hip/hip_runtime.h>
#include <hip/hip_bf16.h>
#include <math.h>

// ---------------- problem constants (match reference) ----------------
#define BATCH      16
#define NGT        32
#define M_ANCH     33600
#define N_TOPK     9
#define NUM_CLS    80
#define BG_IDX     80
#define EPS_F      1e-9f

// levels: 25600, 6400, 1600 (all multiples of 16; boundaries tile-aligned)
// chunking for the fused wmma+topk kernel: 100 tiles (1600 anchors) per chunk
#define TILES_PER_CHUNK 100
#define ANCH_PER_CHUNK  1600
#define CHUNKS_L0 16   // 25600/1600
#define CHUNKS_L1 4    //  6400/1600
#define CHUNKS_L2 1    //  1600/1600
#define NCHUNKS   21   // per batch

__constant__ int c_lvl_start[3] = {0, 25600, 32000};

// ---------------- workspace layout (bytes) ----------------
// partial top-9 values : 512*21*9 f32  = 387072
// partial top-9 indices: 512*21*9 i32  = 387072
// cand_idx             : 512*27  i32  =  55296
// fg_cnt               : 16*33600 i32 = 2150400
// tgt_min              : 16*33600 i32 = 2150400
// total ~5.13 MB
#define OFF_PVAL 0u
#define OFF_PIDX 387072u
#define OFF_CAND 774144u
#define OFF_FG   829440u
#define OFF_TGT  2979840u

// ---------------- output layout (floats) ----------------
#define OUT_LABELS 0u
#define OUT_BBOX   (BATCH * M_ANCH)                       //   537600
#define OUT_SCORES (OUT_BBOX + BATCH * M_ANCH * 4)        //  2688000
#define OUT_FG     (OUT_SCORES + (size_t)BATCH * M_ANCH * NUM_CLS) // 45696000

typedef __attribute__((ext_vector_type(2))) float v2f;
typedef __attribute__((ext_vector_type(8))) float v8f;

__device__ __forceinline__ float box_iou(float4 b1, float4 b2) {
    float lx = fmaxf(b1.x, b2.x), ly = fmaxf(b1.y, b2.y);
    float rx = fminf(b1.z, b2.z), ry = fminf(b1.w, b2.w);
    float iw = fmaxf(rx - lx, 0.0f), ih = fmaxf(ry - ly, 0.0f);
    float inter = iw * ih;
    float a1 = (b1.z - b1.x) * (b1.w - b1.y);
    float a2 = (b2.z - b2.x) * (b2.w - b2.y);
    return inter / (a1 + a2 - inter + EPS_F);
}

// =====================================================================
// K1: fused WMMA distance-score tiles + per-gt top-9 (per chunk).
// score s[g,a] = -2*(cg . ca) + |ca|^2  (same ranking as euclidean dist)
// One wave per (b, chunk). Two V_WMMA_F32_16X16X4_F32 per 16-anchor tile
// (one per 16-gt tile); all 32 lanes own one gt row in the top-k phase.
// =====================================================================
__global__ __launch_bounds__(32)
void k_dist_topk(const float4* __restrict__ anchors,   // (M,4)
                 const float4* __restrict__ gts,       // (B*NGT,4)
                 float* __restrict__ pval,             // (512*21*9)
                 int*   __restrict__ pidx)
{
    const int cid   = blockIdx.x;
    const int chunk = cid % NCHUNKS;
    const int b     = cid / NCHUNKS;
    const int lane  = threadIdx.x;

    int lvl, cil;
    if (chunk < CHUNKS_L0)                 { lvl = 0; cil = chunk; }
    else if (chunk < CHUNKS_L0 + CHUNKS_L1){ lvl = 1; cil = chunk - CHUNKS_L0; }
    else                                   { lvl = 2; cil = chunk - CHUNKS_L0 - CHUNKS_L1; }
    const int aBase = c_lvl_start[lvl] + cil * ANCH_PER_CHUNK;

    // A operands: 16x4 f32, row vector per gt = [-2gx, -2gy, 1, 0]
    // ISA layout: lanes0-15 -> {K0,K1} in v0/v1 (row M=lane); lanes16-31 -> {K2,K3}
    const int r = lane & 15;
    float4 g0 = gts[b * NGT + r];          // gt tile 0, row r
    float4 g1 = gts[b * NGT + 16 + r];     // gt tile 1, row r
    v2f A0, A1;
    A0.x = (lane < 16) ? (-1.0f * (g0.x + g0.z)) : 1.0f;   // K0 : K2
    A0.y = (lane < 16) ? (-1.0f * (g0.y + g0.w)) : 0.0f;   // K1 : K3
    A1.x = (lane < 16) ? (-1.0f * (g1.x + g1.z)) : 1.0f;
    A1.y = (lane < 16) ? (-1.0f * (g1.y + g1.w)) : 0.0f;

    __shared__ float tile[2 * 16 * 16];    // [gtile][row][col]

    float tv[N_TOPK];
    int   ti[N_TOPK];
#pragma unroll
    for (int j = 0; j < N_TOPK; ++j) { tv[j] = __builtin_inff(); ti[j] = -1; }

    // lane owns gt row g = lane; its LDS row (as float4 quads):
    const int myQuad = ((lane < 16) ? (lane * 16) : (256 + (lane - 16) * 16)) >> 2;

    for (int t = 0; t < TILES_PER_CHUNK; ++t) {
        const int a = aBase + t * 16 + r;
        float4 ab = anchors[a];
        float ax = (ab.x + ab.z) * 0.5f, ay = (ab.y + ab.w) * 0.5f;
        float sq = ax * ax + ay * ay;
        // B operand: 4x16 f32, column per anchor = [ax, ay, ax^2+ay^2, 0]^T
        v2f Bv;
        Bv.x = (lane < 16) ? ax : sq;          // K0 row : K2 row
        Bv.y = (lane < 16) ? ay : 0.0f;        // K1 row : K3 row
        v8f cz = {0.f,0.f,0.f,0.f,0.f,0.f,0.f,0.f};
        v8f c0 = __builtin_amdgcn_wmma_f32_16x16x4_f32(
                false, A0, false, Bv, (short)0, cz, false, false);
        v8f c1 = __builtin_amdgcn_wmma_f32_16x16x4_f32(
                false, A1, false, Bv, (short)0, cz, false, false);

        __syncthreads();   // WAR: previous iteration's readers done
        const int rowoff = (lane < 16) ? 0 : 8;
#pragma unroll
        for (int v = 0; v < 8; ++v) {
            tile[(rowoff + v) * 16 + r]       = c0[v];
            tile[256 + (rowoff + v) * 16 + r] = c1[v];
        }
        __syncthreads();

        // all 32 lanes: lane owns gt row 'lane'. Read the row as 4 float4s
        // (ds_load_b128); per-quad wave-uniform pre-filter, then a
        // sequentially-dependent compare-swap (bubble) insert that the
        // compiler cannot re-branchify (each step feeds the next).
        const int idxBase = aBase + t * 16;
#pragma unroll
        for (int k = 0; k < 4; ++k) {
            float4 q = ((const float4*)tile)[myQuad + k];
            float qmin = fminf(fminf(q.x, q.y), fminf(q.z, q.w));
            if (__any(qmin < tv[N_TOPK - 1])) {
                float e[4] = {q.x, q.y, q.z, q.w};
#pragma unroll
                for (int u = 0; u < 4; ++u) {
                    float cv = e[u];
                    int   cI = idxBase + k * 4 + u;
#pragma unroll
                    for (int p = 0; p < N_TOPK; ++p) {
                        float ov = tv[p]; int oi = ti[p];
                        bool take = cv < ov;              // strict: stable ties
                        tv[p] = take ? cv : ov;
                        ti[p] = take ? cI : oi;
                        cv    = take ? ov : cv;           // carry the max down
                        cI    = take ? oi : cI;
                    }
                }
            }
        }
    }

    const int gi   = b * NGT + lane;       // lane == gt row
    const int base = (gi * NCHUNKS + chunk) * N_TOPK;
#pragma unroll
    for (int j = 0; j < N_TOPK; ++j) { pval[base + j] = tv[j]; pidx[base + j] = ti[j]; }
}

// =====================================================================
// K2: merge per-chunk sorted top-9 lists into final 9 per (b,g,level).
// =====================================================================
__global__ void k_merge(const float* __restrict__ pval,
                        const int*   __restrict__ pidx,
                        int*         __restrict__ cand)   // (512*27)
{
    const int t = blockIdx.x * blockDim.x + threadIdx.x;
    if (t >= BATCH * NGT * 3) return;
    const int gi  = t / 3;
    const int lvl = t % 3;
    const int cs  = (lvl == 0) ? 0 : (lvl == 1 ? CHUNKS_L0 : CHUNKS_L0 + CHUNKS_L1);
    const int nc  = (lvl == 0) ? CHUNKS_L0 : (lvl == 1 ? CHUNKS_L1 : CHUNKS_L2);

    int ptr[CHUNKS_L0];
    for (int c = 0; c < CHUNKS_L0; ++c) ptr[c] = 0;

    for (int pick = 0; pick < N_TOPK; ++pick) {
        float bestV = __builtin_inff(); int bestI = 0x7fffffff; int bestC = 0;
        for (int c = 0; c < nc; ++c) {
            if (ptr[c] < N_TOPK) {
                int off = (gi * NCHUNKS + cs + c) * N_TOPK + ptr[c];
                float v = pval[off]; int ix = pidx[off];
                if (v < bestV || (v == bestV && ix < bestI)) { bestV = v; bestI = ix; bestC = c; }
            }
        }
        ptr[bestC]++;
        cand[gi * 27 + lvl * N_TOPK + pick] = bestI;
    }
}

// =====================================================================
// K3: per (b,g): IoU of 27 candidates -> thr = mean + std(ddof=1);
// positive = iou>thr && anchor-center strictly inside gt && mask_gt.
// Scatter: fg_cnt += 1 (atomicAdd), tgt_min = min(g) (atomicMin).
// =====================================================================
__global__ void k_thr_flags(const float4* __restrict__ anchors,
                            const float4* __restrict__ gts,
                            const float*  __restrict__ mask_gt,   // (B*NGT)
                            const int*    __restrict__ cand,
                            int* __restrict__ fg_cnt,             // (B*M)
                            int* __restrict__ tgt_min)
{
    const int gi = blockIdx.x * blockDim.x + threadIdx.x;
    if (gi >= BATCH * NGT) return;
    const int b = gi / NGT;
    const int g = gi % NGT;
    if (mask_gt[gi] == 0.0f) return;   // masked gt contributes nothing

    const float4 gb = gts[gi];
    float iou[27]; int ci[27];
    float sum = 0.0f;
#pragma unroll
    for (int j = 0; j < 27; ++j) {
        ci[j]  = cand[gi * 27 + j];
        iou[j] = box_iou(gb, anchors[ci[j]]);
        sum += iou[j];
    }
    const float mean = sum * (1.0f / 27.0f);
    float var = 0.0f;
#pragma unroll
    for (int j = 0; j < 27; ++j) { float d = iou[j] - mean; var += d * d; }
    const float thr = mean + sqrtf(var * (1.0f / 26.0f));   // ddof=1

#pragma unroll
    for (int j = 0; j < 27; ++j) {
        if (iou[j] > thr) {
            float4 ab = anchors[ci[j]];
            float px = (ab.x + ab.z) * 0.5f, py = (ab.y + ab.w) * 0.5f;
            float mind = fminf(fminf(px - gb.x, py - gb.y),
                               fminf(gb.z - px, gb.w - py));
            if (mind > EPS_F) {
                const int off = b * M_ANCH + ci[j];
                atomicAdd(&fg_cnt[off], 1);
                atomicMin(&tgt_min[off], g);
            }
        }
    }
}

// =====================================================================
// K4: per (b,a) resolve target gt + write all outputs.
// fg==1 -> tgt from atomicMin (first positive g == argmax of one-hot).
// fg>1  -> tgt = argmax_g IoU(gt[b,g], anchor[a])  (first-max, all gts).
// =====================================================================
__global__ void k_resolve(const float4* __restrict__ anchors,
                          const float4* __restrict__ gts,
                          const int*    __restrict__ gt_labels,   // (B*NGT)
                          const float4* __restrict__ preds,       // (B*M,4)
                          const int* __restrict__ fg_cnt,
                          const int* __restrict__ tgt_min,
                          float* __restrict__ out)
{
    const int tid = blockIdx.x * blockDim.x + threadIdx.x;
    if (tid >= BATCH * M_ANCH) return;
    const int b = tid / M_ANCH;
    const int a = tid % M_ANCH;

    const int fg = fg_cnt[tid];
    int tgt; bool fgb;
    if (fg == 0)      { tgt = 0; fgb = false; }
    else if (fg == 1) { tgt = tgt_min[tid]; fgb = true; }
    else {
        // ambiguous anchor: keep gt with max overlap (first max over ALL gts)
        float4 ab = anchors[a];
        float best = -1.0f; tgt = 0;
        for (int g = 0; g < NGT; ++g) {
            float v = box_iou(gts[b * NGT + g], ab);
            if (v > best) { best = v; tgt = g; }
        }
        fgb = true;
    }

    const int label = fgb ? gt_labels[b * NGT + tgt] : BG_IDX;
    out[OUT_LABELS + tid] = (float)label;

    const float4 tb = gts[b * NGT + tgt];   // taken regardless of fg (reference)
    ((float4*)(out + OUT_BBOX))[tid] = tb;  // global_store_b128

    out[OUT_FG + tid] = fgb ? 1.0f : 0.0f;

    if (fgb) {   // single nonzero in the 80-wide one-hot score row
        float s = box_iou(tb, preds[tid]);
        out[OUT_SCORES + (size_t)tid * NUM_CLS + label] = s;
    }
}

// =====================================================================
extern "C" void kernel_launch(void* const* d_in, const int* in_sizes, int n_in,
                              void* d_out, int out_size, void* d_ws, size_t ws_size,
                              hipStream_t stream)
{
    const float4* anchors = (const float4*)d_in[0];   // (M,4)
    // d_in[1] = n_level_bboxes (compile-time constants here)
    const float4* gts     = (const float4*)d_in[2];   // (B,NGT,4)
    const int*    labels  = (const int*)  d_in[3];    // (B,NGT,1)
    const float*  mask_gt = (const float*)d_in[4];    // (B,NGT,1)
    const float4* preds   = (const float4*)d_in[5];   // (B,M,4)

    char* ws = (char*)d_ws;
    float* pval  = (float*)(ws + OFF_PVAL);
    int*   pidx  = (int*)  (ws + OFF_PIDX);
    int*   cand  = (int*)  (ws + OFF_CAND);
    int*   fgcnt = (int*)  (ws + OFF_FG);
    int*   tgtmn = (int*)  (ws + OFF_TGT);
    float* out   = (float*)d_out;

    // init (graph-capture-safe async memsets): fg=0, tgt=+big, scores=0
    hipMemsetAsync(fgcnt, 0,    (size_t)BATCH * M_ANCH * 4, stream);
    hipMemsetAsync(tgtmn, 0x7f, (size_t)BATCH * M_ANCH * 4, stream);
    hipMemsetAsync(out + OUT_SCORES, 0,
                   (size_t)BATCH * M_ANCH * NUM_CLS * sizeof(float), stream);

    // K1: 16 batches * 21 chunks, one wave32 each (2 WMMAs / anchor tile)
    k_dist_topk<<<BATCH * NCHUNKS, 32, 0, stream>>>(anchors, gts, pval, pidx);

    // K2: merge chunk partials -> 27 candidates per (b,g)
    {
        int n = BATCH * NGT * 3;
        k_merge<<<(n + 255) / 256, 256, 0, stream>>>(pval, pidx, cand);
    }

    // K3: threshold + positivity flags (sparse scatter via atomics)
    {
        int n = BATCH * NGT;
        k_thr_flags<<<(n + 255) / 256, 256, 0, stream>>>(anchors, gts, mask_gt,
                                                         cand, fgcnt, tgtmn);
    }

    // K4: per-anchor resolve + write labels/bboxes/scores/fg
    {
        int n = BATCH * M_ANCH;
        k_resolve<<<(n + 255) / 256, 256, 0, stream>>>(anchors, gts, labels, preds,
                                                       fgcnt, tgtmn, out);
    }
}